// SelfAttentionVectorLayer_2388001817350
// MI455X (gfx1250) — compile-verified
//
#include <hip/hip_runtime.h>

// ---------------------------------------------------------------------------
// Problem constants (match reference): B=4, S=4096, H=1024, D=64
// out = Q (K^T V) / sqrt(D)   (no softmax -> associativity refactor)
// ---------------------------------------------------------------------------
#define BATCH 4
#define SEQ   4096
#define HID   1024
#define DHEAD 64
#define BS    (BATCH * SEQ)          // 16384 rows
#define SPLITS 64                    // split-K chunks over SEQ for K^T V
#define CHUNK (SEQ / SPLITS)         // 64 s-values per chunk
#define KC    32                     // K-chunk staged in LDS per block
#define NCHUNK (HID / KC)            // 32 chunks
#define NPAIR (KC / 2)               // 16 k-pairs per chunk
#define PSTRIDE 208                  // pair-row stride in float2 (192 + 16 pad)
                                     // 416 dwords = 32 mod 64 banks -> the two
                                     // lane-halves hit disjoint bank halves

typedef __attribute__((ext_vector_type(2))) float v2f;
typedef __attribute__((ext_vector_type(8))) float v8f;

// V_WMMA_F32_16X16X4_F32: D(16x16 f32) = A(16x4 f32) * B(4x16 f32) + C
__device__ __forceinline__ v8f wmma_f32(v2f a, v2f b, v8f c) {
  return __builtin_amdgcn_wmma_f32_16x16x4_f32(false, a, false, b, (short)0, c,
                                               false, false);
}

__device__ __forceinline__ v8f vzero8() {
  v8f z = {0.f, 0.f, 0.f, 0.f, 0.f, 0.f, 0.f, 0.f};
  return z;
}

// gfx1250 async DMA: copies 4 bytes from the global address gaddr to the LDS byte offset lds_off; tracked by ASYNCcnt.
// 4B granularity lets the copy itself perform the pair-interleave permute.
__device__ __forceinline__ void async_copy_b32(uint32_t lds_off,
                                               const float* gaddr) {
  asm volatile("global_load_async_to_lds_b32 %0, %1, off"
               :
               : "v"(lds_off), "v"(gaddr)
               : "memory");
}

__device__ __forceinline__ void wait_async0() {
  asm volatile("s_wait_asynccnt 0" ::: "memory");
}

// Stage one K-chunk of Wq|Wk|Wv into LDS in WMMA pair-major layout:
//   float2 at [p][wcol] = { W[k0+2p, c], W[k0+2p+1, c] },  wcol = w*64 + c.
// 3*32*64 = 6144 elements -> 24 async b32 copies per thread, coalesced reads.
__device__ __forceinline__ void stage_weights(const float* __restrict__ Wq,
                                              const float* __restrict__ Wk,
                                              const float* __restrict__ Wv,
                                              int k0, uint32_t lds_byte_base,
                                              int tid) {
#pragma unroll
  for (int i = 0; i < 24; ++i) {
    const int f = i * 256 + tid;
    const int c  = f & 63;           // column within one weight matrix
    const int kk = (f >> 6) & 31;    // k-row within chunk
    const int w  = f >> 11;          // 0=Q, 1=K, 2=V (uniform per wave)
    const float* g = (w == 0 ? Wq : (w == 1 ? Wk : Wv)) +
                     (size_t)(k0 + kk) * DHEAD + c;
    const uint32_t loff =
        lds_byte_base +
        (uint32_t)(((((kk >> 1) * PSTRIDE) + w * 64 + c) * 2 + (kk & 1)) * 4);
    async_copy_b32(loff, g);
  }
}

// ---------------------------------------------------------------------------
// Kernel 1: fused QKV projection.  [BS,H] @ [H,64] x3  -> q,k,v [BS,64]
// 8 waves/block, wave -> 16 rows x 192 cols (12 wmma accumulators).
// Weights double-buffered in LDS (pair-major) via async global->LDS DMA;
// each B fragment is a single conflict-free aligned ds_load_b64.
// grid: BS/(16*8) = 128 blocks of 256 threads.
// ---------------------------------------------------------------------------
__global__ __launch_bounds__(256) void qkv_proj_kernel(
    const float* __restrict__ x,   // [BS, H]
    const float* __restrict__ Wk,  // [H, D]
    const float* __restrict__ Wq,  // [H, D]
    const float* __restrict__ Wv,  // [H, D]
    float* __restrict__ qb, float* __restrict__ kb, float* __restrict__ vb) {
  __shared__ __align__(16) v2f ldsw[2][NPAIR * PSTRIDE];  // 52 KB

  const int lane = threadIdx.x & 31;
  const int wave = threadIdx.x >> 5;
  const int m0 = (blockIdx.x * 8 + wave) * 16;       // output row base
  const int arow = lane & 15;
  const int koff = (lane >> 4) << 1;                 // 0 or 2
  const int bcol = lane & 15;

  const uint32_t lds_base0 = (uint32_t)(uintptr_t)&ldsw[0][0];
  const uint32_t lds_base1 = (uint32_t)(uintptr_t)&ldsw[1][0];

  v8f accq[4], acck[4], accv[4];
#pragma unroll
  for (int i = 0; i < 4; ++i) {
    accq[i] = vzero8(); acck[i] = vzero8(); accv[i] = vzero8();
  }

  const float* xrow = x + (size_t)(m0 + arow) * HID;

  // Prologue: stage chunk 0 into buffer 0.
  stage_weights(Wq, Wk, Wv, 0, lds_base0, threadIdx.x);

  for (int c = 0; c < NCHUNK; ++c) {
    wait_async0();          // this wave's copies for chunk c have landed in LDS
    __syncthreads();        // all waves' copies have landed

    if (c + 1 < NCHUNK)     // overlap next chunk's DMA with this chunk's WMMAs
      stage_weights(Wq, Wk, Wv, (c + 1) * KC,
                    ((c + 1) & 1) ? lds_base1 : lds_base0, threadIdx.x);

    const v2f* lp = &ldsw[c & 1][0];
    const float* xr = xrow + c * KC;

#pragma unroll
    for (int ks = 0; ks < KC / 4; ++ks) {
      const v2f a = *(const v2f*)(xr + ks * 4 + koff);   // A[m, kk..kk+1]
      const int p = 2 * ks + (lane >> 4);                // k-pair index
      const int base = p * PSTRIDE + bcol;
#pragma unroll
      for (int nt = 0; nt < 4; ++nt) {
        const int nc = base + nt * 16;
        accq[nt] = wmma_f32(a, lp[nc],       accq[nt]);  // Q cols 0..63
        acck[nt] = wmma_f32(a, lp[nc + 64],  acck[nt]);  // K cols 64..127
        accv[nt] = wmma_f32(a, lp[nc + 128], accv[nt]);  // V cols 128..191
      }
    }
    __syncthreads();        // buffer (c&1) free for reuse at chunk c+2
  }

  // C/D layout: vgpr j, lanes 0-15 -> M=j, lanes 16-31 -> M=j+8; N = lane&15
  const int mhalf = (lane >> 4) << 3;
#pragma unroll
  for (int nt = 0; nt < 4; ++nt) {
#pragma unroll
    for (int j = 0; j < 8; ++j) {
      const size_t idx =
          (size_t)(m0 + mhalf + j) * DHEAD + nt * 16 + (lane & 15);
      qb[idx] = accq[nt][j];
      kb[idx] = acck[nt][j];
      vb[idx] = accv[nt][j];
    }
  }
}

// ---------------------------------------------------------------------------
// Kernel 2: M[b] += K[b]^T V[b]  (64x64 per batch), split-K over SEQ,
// f32 atomic reduction into Mbuf.  grid: BATCH*SPLITS blocks of 256 threads.
// Each of the 8 waves computes 2 of the 16 16x16 output tiles.
// ---------------------------------------------------------------------------
__global__ __launch_bounds__(256) void ktv_kernel(
    const float* __restrict__ kb, const float* __restrict__ vb,
    float* __restrict__ Mbuf) {
  const int lane = threadIdx.x & 31;
  const int wave = threadIdx.x >> 5;
  const int b = blockIdx.x / SPLITS;
  const int s0 = (blockIdx.x % SPLITS) * CHUNK;
  const int arow = lane & 15;
  const int koff = (lane >> 4) << 1;
  const int bcol = lane & 15;
  const int mhalf = (lane >> 4) << 3;

  const float* kmat = kb + (size_t)b * SEQ * DHEAD;
  const float* vmat = vb + (size_t)b * SEQ * DHEAD;
  float* Mb = Mbuf + (size_t)b * DHEAD * DHEAD;

#pragma unroll
  for (int t = 0; t < 2; ++t) {
    const int tile = wave * 2 + t;                   // 0..15
    const int ar0 = (tile >> 2) * 16;                // row base (k-feature i)
    const int bc0 = (tile & 3) * 16;                 // col base (v-feature j)
    v8f acc = vzero8();
    for (int s = s0; s < s0 + CHUNK; s += 4) {
      // A[i, s'] = k[s', i]  (transposed gather), B[s', j] = v[s', j]
      v2f a, bbv;
      a.x   = kmat[(size_t)(s + koff)     * DHEAD + ar0 + arow];
      a.y   = kmat[(size_t)(s + koff + 1) * DHEAD + ar0 + arow];
      bbv.x = vmat[(size_t)(s + koff)     * DHEAD + bc0 + bcol];
      bbv.y = vmat[(size_t)(s + koff + 1) * DHEAD + bc0 + bcol];
      acc = wmma_f32(a, bbv, acc);
    }
#pragma unroll
    for (int j = 0; j < 8; ++j) {
      atomicAdd(&Mb[(ar0 + mhalf + j) * DHEAD + bc0 + (lane & 15)], acc[j]);
    }
  }
}

// ---------------------------------------------------------------------------
// Kernel 3: out = Q @ M[b] * (1/sqrt(D)).  [BS,64] @ [64,64].
// grid: 128 blocks of 256 threads; wave -> 16 rows x 64 cols.
// ---------------------------------------------------------------------------
__global__ __launch_bounds__(256) void qm_kernel(
    const float* __restrict__ qb, const float* __restrict__ Mbuf,
    float* __restrict__ out) {
  const int lane = threadIdx.x & 31;
  const int wave = threadIdx.x >> 5;
  const int m0 = (blockIdx.x * 8 + wave) * 16;       // rows; never straddles batch
  const int b = m0 / SEQ;
  const int arow = lane & 15;
  const int koff = (lane >> 4) << 1;
  const int bcol = lane & 15;
  const int mhalf = (lane >> 4) << 3;

  const float* qrow = qb + (size_t)(m0 + arow) * DHEAD;
  const float* Mb = Mbuf + (size_t)b * DHEAD * DHEAD;

  v8f acc[4];
#pragma unroll
  for (int i = 0; i < 4; ++i) acc[i] = vzero8();

#pragma unroll
  for (int kk = 0; kk < DHEAD; kk += 4) {
    const v2f a = *(const v2f*)(qrow + kk + koff);
    const int r0 = (kk + koff) * DHEAD;
    const int r1 = r0 + DHEAD;
#pragma unroll
    for (int nt = 0; nt < 4; ++nt) {
      v2f bbm;
      bbm.x = Mb[r0 + nt * 16 + bcol];
      bbm.y = Mb[r1 + nt * 16 + bcol];
      acc[nt] = wmma_f32(a, bbm, acc[nt]);
    }
  }

  const float scale = 0.125f;                        // 1/sqrt(64)
#pragma unroll
  for (int nt = 0; nt < 4; ++nt) {
#pragma unroll
    for (int j = 0; j < 8; ++j) {
      out[(size_t)(m0 + mhalf + j) * DHEAD + nt * 16 + (lane & 15)] =
          acc[nt][j] * scale;
    }
  }
}

// ---------------------------------------------------------------------------
// Host launcher. Inputs: x, Wk, Wq, Wv (fp32). Output: [B,S,D] fp32.
// Workspace layout: q | k | v (4 MB each) | M (64 KB).
// ---------------------------------------------------------------------------
extern "C" void kernel_launch(void* const* d_in, const int* in_sizes, int n_in,
                              void* d_out, int out_size, void* d_ws,
                              size_t ws_size, hipStream_t stream) {
  (void)in_sizes; (void)n_in; (void)out_size; (void)ws_size;
  const float* x  = (const float*)d_in[0];
  const float* Wk = (const float*)d_in[1];
  const float* Wq = (const float*)d_in[2];
  const float* Wv = (const float*)d_in[3];
  float* out = (float*)d_out;

  const size_t proj_elems = (size_t)BS * DHEAD;      // 1,048,576
  float* qb = (float*)d_ws;
  float* kb = qb + proj_elems;
  float* vb = kb + proj_elems;
  float* Mb = vb + proj_elems;                       // BATCH*64*64 floats

  hipMemsetAsync(Mb, 0, (size_t)BATCH * DHEAD * DHEAD * sizeof(float), stream);

  qkv_proj_kernel<<<BS / 128, 256, 0, stream>>>(x, Wk, Wq, Wv, qb, kb, vb);
  ktv_kernel<<<BATCH * SPLITS, 256, 0, stream>>>(kb, vb, Mb);
  qm_kernel<<<BS / 128, 256, 0, stream>>>(qb, Mb, out);
}